// ChamferDistanceLoss_13314398618340
// MI455X (gfx1250) — compile-verified
//
#include <hip/hip_runtime.h>
#include <math.h>

typedef __attribute__((ext_vector_type(2))) float v2f;
typedef __attribute__((ext_vector_type(8))) float v8f;

#define N_PTS 4096
#define BATCH 16
#define ROWS_PER_WG 128
#define THREADS 256

// ---------------------------------------------------------------------------
// Kernel 1: init workspace minima to +inf (bit pattern 0x7F800000)
// ---------------------------------------------------------------------------
__global__ void chamfer_init_ws(unsigned int* __restrict__ ws) {
    int idx = blockIdx.x * blockDim.x + threadIdx.x;
    if (idx < 2 * BATCH * N_PTS) ws[idx] = 0x7F800000u;
}

// ---------------------------------------------------------------------------
// Kernel 2: tiled pairwise sq-dist via V_WMMA_F32_16X16X4_F32 + min reductions
//   blockIdx.y = batch, blockIdx.x = 128-row template strip (32 per batch)
//   256 threads = 8 waves; each wave owns 16 template rows, loops 256 j-tiles
// ---------------------------------------------------------------------------
__global__ __launch_bounds__(THREADS)
void chamfer_tiles(const float* __restrict__ tmpl,
                   const float* __restrict__ src,
                   float* __restrict__ rowmin_g,          // [16][4096] float
                   unsigned int* __restrict__ colmin_g) { // [16][4096] f32-as-u32
    // Source staged as (x, y, z, |s|^2): one ds_load_b128 per tile, norm
    // precomputed once. 16B stride -> banks 4j mod 64: conflict-free.
    __shared__ float4       lds_src[N_PTS];         // 64 KB
    __shared__ unsigned int lds_colmin[N_PTS];      // 16 KB

    const int b    = blockIdx.y;
    const int tid  = threadIdx.x;
    const int lane = tid & 31;
    const int wave = tid >> 5;
    const int half = lane >> 4;   // 0: lanes 0-15 (rows 0-7), 1: lanes 16-31 (rows 8-15)
    const int l16  = lane & 15;

    // Stage source cloud into LDS with precomputed squared norm.
    const float* srcB = src + (size_t)b * N_PTS * 3;
    for (int k = tid; k < N_PTS; k += THREADS) {
        const float x = srcB[k * 3 + 0];
        const float y = srcB[k * 3 + 1];
        const float z = srcB[k * 3 + 2];
        lds_src[k] = make_float4(x, y, z, x * x + y * y + z * z);
    }
    for (int k = tid; k < N_PTS; k += THREADS) lds_colmin[k] = 0x7F800000u;
    __syncthreads();

    // A operand: A'[i] = (-2tx, -2ty, -2tz, |t|^2), row i = iwb + l16.
    // Layout: lanes 0-15 hold K=0 (v0) and K=1 (v1); lanes 16-31 hold K=2,K=3.
    const int iwb  = blockIdx.x * ROWS_PER_WG + wave * 16;
    const int irow = iwb + l16;
    const float* tp = tmpl + ((size_t)b * N_PTS + irow) * 3;
    const float tx = tp[0], ty = tp[1], tz = tp[2];
    const float a2 = tx * tx + ty * ty + tz * tz;
    v2f A;
    A.x = half ? (-2.0f * tz) : (-2.0f * tx);
    A.y = half ? a2           : (-2.0f * ty);

    float rmin[8];
#pragma unroll
    for (int v = 0; v < 8; ++v) rmin[v] = __builtin_inff();

    // Incremented pointers: DS addresses strength-reduce to adds.
    const float4* sp = lds_src + l16;          // this lane's source point
    unsigned int* cp = lds_colmin + l16;       // this lane's column-min slot

#pragma unroll 2
    for (int jt = 0; jt < N_PTS / 16; ++jt) {
        const float4 s = *sp;                  // ds_load_b128: x, y, z, |s|^2
        const float  b2 = s.w;

        // B operand: B'[:,j] = (sx, sy, sz, 1)^T.
        // VGPR0 = rows K=0 (lanes 0-15) / K=2 (lanes 16-31); VGPR1 = K=1 / K=3.
        v2f B;
        B.x = half ? s.z  : s.x;
        B.y = half ? 1.0f : s.y;

        v8f C = {};
        // D[i][j] = |t_i|^2 - 2 t_i . s_j   (16x16 tile, one wmma)
        C = __builtin_amdgcn_wmma_f32_16x16x4_f32(
                /*neg_a=*/false, A, /*neg_b=*/false, B,
                /*c_mod=*/(short)0, C, /*reuse_a=*/false, /*reuse_b=*/false);

        // Full M = C + |s_j|^2 (unclamped; clamp commutes with min).
        float cmin = __builtin_inff();
#pragma unroll
        for (int v = 0; v < 8; ++v) {
            const float m = C[v] + b2;
            rmin[v] = fminf(rmin[v], m);   // running row minima (per lane/column)
            cmin = fminf(cmin, m);         // this half's 8-row column minimum
        }
        // Both 16-lane halves atomically min into the SAME column slot:
        // the LDS atomic unit merges rows 0-7 and 8-15 — no shuffle, no branch.
        // Clamp before bit-cast so uint ordering == float ordering (all >= 0).
        atomicMin(cp, __float_as_uint(fmaxf(cmin, 0.0f)));  // ds_min_u32

        sp += 16;
        cp += 16;
    }

    // Row minima: reduce across the 16 lanes of each half (cols j%16 covered
    // per lane across the jt loop). xor masks 1,2,4,8 stay within each half.
#pragma unroll
    for (int v = 0; v < 8; ++v) {
        float r = rmin[v];
        r = fminf(r, __shfl_xor(r, 1, 32));
        r = fminf(r, __shfl_xor(r, 2, 32));
        r = fminf(r, __shfl_xor(r, 4, 32));
        r = fminf(r, __shfl_xor(r, 8, 32));
        if (l16 == 0) {
            const int i = iwb + v + half * 8;   // this row owned by this wave only
            rowmin_g[b * N_PTS + i] = fmaxf(r, 0.0f);
        }
    }

    // Publish WG-local column minima to global (min over all template strips).
    // All stored values are clamped >= 0, so uint atomic-min is order-correct.
    __syncthreads();
    for (int k = tid; k < N_PTS; k += THREADS) {
        atomicMin(&colmin_g[b * N_PTS + k], lds_colmin[k]); // global_atomic_min_u32
    }
}

// ---------------------------------------------------------------------------
// Kernel 3: final scalar = ( mean(sqrt(colmin)) + mean(sqrt(rowmin)) ) / 2
// ---------------------------------------------------------------------------
__global__ __launch_bounds__(THREADS)
void chamfer_finalize(const float* __restrict__ rowmin_g,
                      const unsigned int* __restrict__ colmin_g,
                      float* __restrict__ out) {
    __shared__ float s1[THREADS], s2[THREADS];
    const int tid = threadIdx.x;
    float sum_r = 0.0f, sum_c = 0.0f;
    for (int k = tid; k < BATCH * N_PTS; k += THREADS) {
        sum_r += sqrtf(rowmin_g[k]);
        sum_c += sqrtf(__uint_as_float(colmin_g[k]));
    }
    s1[tid] = sum_r;
    s2[tid] = sum_c;
    __syncthreads();
    for (int s = THREADS / 2; s > 0; s >>= 1) {
        if (tid < s) { s1[tid] += s1[tid + s]; s2[tid] += s2[tid + s]; }
        __syncthreads();
    }
    if (tid == 0) {
        const float inv = 1.0f / (float)(BATCH * N_PTS);
        out[0] = 0.5f * (s1[0] * inv + s2[0] * inv);
    }
}

// ---------------------------------------------------------------------------
extern "C" void kernel_launch(void* const* d_in, const int* in_sizes, int n_in,
                              void* d_out, int out_size, void* d_ws, size_t ws_size,
                              hipStream_t stream) {
    const float* tmpl = (const float*)d_in[0];   // [16,4096,3] f32
    const float* src  = (const float*)d_in[1];   // [16,4096,3] f32

    float*        rowmin = (float*)d_ws;                                   // [16*4096]
    unsigned int* colmin = (unsigned int*)((char*)d_ws +
                                           (size_t)BATCH * N_PTS * sizeof(float));

    const int total = 2 * BATCH * N_PTS;
    chamfer_init_ws<<<(total + THREADS - 1) / THREADS, THREADS, 0, stream>>>(
        (unsigned int*)d_ws);

    dim3 grid(N_PTS / ROWS_PER_WG, BATCH);   // 32 x 16 = 512 workgroups
    chamfer_tiles<<<grid, THREADS, 0, stream>>>(tmpl, src, rowmin, colmin);

    chamfer_finalize<<<1, THREADS, 0, stream>>>(rowmin, colmin, (float*)d_out);
}